// RegionLoss_83399674954428
// MI455X (gfx1250) — compile-verified
//
#include <hip/hip_runtime.h>
#include <math.h>

#define NB 64
#define NA 5
#define NC 80
#define NH 38
#define NW 38
#define MAXT 50
#define CH (5 + NC)              // 85
#define HW (NH * NW)             // 1444
#define CELLS (NB * NA * HW)     // 462080 = 1805 * 256 exactly
#define NBLK_A (CELLS / 256)     // 1805

__device__ __constant__ float c_aw[NA] = {0.57273f, 1.87446f, 3.33843f, 7.88282f, 9.77052f};
__device__ __constant__ float c_ah[NA] = {0.677385f, 2.06253f, 5.47434f, 3.52778f, 9.16828f};

typedef __attribute__((ext_vector_type(2))) float v2f;
typedef __attribute__((ext_vector_type(8))) float v8f;

__device__ __forceinline__ float sigmoidf_(float v) { return 1.0f / (1.0f + __expf(-v) * 0.0f + expf(-v)); }

__device__ __forceinline__ float iou_box(float x1, float y1, float w1, float h1,
                                         float x2, float y2, float w2, float h2) {
  float iw = fminf(x1 + 0.5f * w1, x2 + 0.5f * w2) - fmaxf(x1 - 0.5f * w1, x2 - 0.5f * w2);
  iw = fmaxf(iw, 0.0f);
  float ih = fminf(y1 + 0.5f * h1, y2 + 0.5f * h2) - fmaxf(y1 - 0.5f * h1, y2 - 0.5f * h2);
  ih = fmaxf(ih, 0.0f);
  float inter = iw * ih;
  float uni = w1 * h1 + w2 * h2 - inter;
  return inter / fmaxf(uni, 1e-12f);
}

// Exact f32 sum of all 32 lane values using V_WMMA_F32_16X16X4_F32.
// A[m][0]=v[m] (lanes 0-15), A[m][2]=v[m+16] (lanes 16-31), B = ones.
// D[m][n] = v[m] + v[m+16] for all n. Lane L holds rows {0..7} (L<16) or
// {8..15} (L>=16) of column L%16 in its 8 accumulator VGPRs; summing those
// plus one cross-half shuffle yields the full 32-lane sum in every lane.
__device__ __forceinline__ float wave_sum_wmma(float v) {
  v2f a; a[0] = v;    a[1] = 0.0f;
  v2f b; b[0] = 1.0f; b[1] = 1.0f;
  v8f c = {0.f, 0.f, 0.f, 0.f, 0.f, 0.f, 0.f, 0.f};
  v8f d = __builtin_amdgcn_wmma_f32_16x16x4_f32(false, a, false, b, (short)0, c, false, false);
  float t = d[0] + d[1] + d[2] + d[3] + d[4] + d[5] + d[6] + d[7];
  t += __shfl_xor(t, 16, 32);
  return t;
}

__device__ __forceinline__ float block_sum_256(float v, float* s8) {
  float w = wave_sum_wmma(v);               // all waves execute: EXEC all ones
  int wid = threadIdx.x >> 5, lane = threadIdx.x & 31;
  if (lane == 0) s8[wid] = w;
  __syncthreads();
  float s = 0.0f;
  if (threadIdx.x == 0)
    for (int k = 0; k < 8; ++k) s += s8[k];
  return s;                                 // valid on thread 0 only
}

__global__ void kinit(float* tconf_ovr) {
  int idx = blockIdx.x * 256 + threadIdx.x;
  if (idx < CELLS) tconf_ovr[idx] = -1.0f;
}

__global__ __launch_bounds__(128) void ktargets(const float* __restrict__ in,
                                                const float* __restrict__ tgt,
                                                float* __restrict__ tconf_ovr,
                                                float* __restrict__ gtbox,
                                                int* __restrict__ nvalid,
                                                float* __restrict__ partT) {
  int b = blockIdx.x;
  int tid = threadIdx.x;
  __shared__ int s_win[MAXT], s_a[MAXT], s_gi[MAXT], s_gj[MAXT], s_gc[MAXT];
  __shared__ float s_gx[MAXT], s_gy[MAXT], s_gw[MAXT], s_gh[MAXT];
  __shared__ float s_red[128];

  if (tid == 0) {
    const float* tb = tgt + (size_t)b * MAXT * 5;
    int validf = 1;
    int nv = MAXT;
    for (int t = 0; t < MAXT; ++t) {
      float gcls = tb[t * 5 + 0];
      float txn = tb[t * 5 + 1];
      float tyn = tb[t * 5 + 2];
      float twn = tb[t * 5 + 3];
      float thn = tb[t * 5 + 4];
      if (txn == 0.0f) validf = 0;
      if (!validf && nv == MAXT) nv = t;
      float gx = txn * NW, gy = tyn * NH, gw = twn * NW, gh = thn * NH;
      // best anchor (boxes centered at origin): first max wins ties
      int bestn = 0; float bestiou = -1.0f;
      for (int a = 0; a < NA; ++a) {
        float iou = iou_box(0.f, 0.f, gw, gh, 0.f, 0.f, c_aw[a], c_ah[a]);
        if (iou > bestiou) { bestiou = iou; bestn = a; }
      }
      int gi = (int)gx, gj = (int)gy;     // truncation == floor for >=0
      s_a[t] = bestn; s_gi[t] = gi; s_gj[t] = gj;
      s_gx[t] = gx; s_gy[t] = gy; s_gw[t] = gw; s_gh[t] = gh;
      s_gc[t] = (int)gcls;
      s_win[t] = (validf && gi >= 0 && gi < NW && gj >= 0 && gj < NH) ? 1 : 0;
      float* gb = gtbox + ((size_t)b * MAXT + t) * 4;
      gb[0] = gx; gb[1] = gy; gb[2] = gw; gb[3] = gh;
    }
    nvalid[b] = nv;
    // duplicate scatter resolution: last update wins (in-order scatter)
    for (int t = 0; t < MAXT; ++t) {
      if (!s_win[t]) continue;
      for (int u = t + 1; u < MAXT; ++u) {
        if (s_win[u] && s_a[u] == s_a[t] && s_gi[u] == s_gi[t] && s_gj[u] == s_gj[t]) {
          s_win[t] = 0; break;
        }
      }
    }
  }
  __syncthreads();

  float myloss = 0.0f;   // accumulated on thread 0 only
  for (int t = 0; t < MAXT; ++t) {
    if (!s_win[t]) continue;               // uniform (shared) -> no divergence
    int a = s_a[t], gi = s_gi[t], gj = s_gj[t];
    size_t base = ((size_t)(b * NA + a) * CH) * HW + (size_t)gj * NW + gi;

    // log-softmax over 80 class logits at the winner cell
    float logit = -1e30f;
    if (tid < NC) logit = in[base + (size_t)(5 + tid) * HW];
    s_red[tid] = logit;
    __syncthreads();
    for (int off = 64; off > 0; off >>= 1) {
      if (tid < off) s_red[tid] = fmaxf(s_red[tid], s_red[tid + off]);
      __syncthreads();
    }
    float mx = s_red[0];
    __syncthreads();
    s_red[tid] = (tid < NC) ? expf(logit - mx) : 0.0f;
    __syncthreads();
    for (int off = 64; off > 0; off >>= 1) {
      if (tid < off) s_red[tid] += s_red[tid + off];
      __syncthreads();
    }
    float sumexp = s_red[0];
    __syncthreads();

    if (tid == 0) {
      int gc = s_gc[t]; gc = gc < 0 ? 0 : (gc > NC - 1 ? NC - 1 : gc);
      float logit_g = in[base + (size_t)(5 + gc) * HW];
      float l_cls = (mx + logf(sumexp)) - logit_g;   // = -log_softmax[gc]

      float in0 = in[base];
      float in1 = in[base + HW];
      float in2 = in[base + 2 * (size_t)HW];
      float in3 = in[base + 3 * (size_t)HW];
      float sx = 1.0f / (1.0f + expf(-in0));
      float sy = 1.0f / (1.0f + expf(-in1));
      float txv = s_gx[t] - (float)gi;
      float tyv = s_gy[t] - (float)gj;
      float twv = logf(s_gw[t] / c_aw[a]);
      float thv = logf(s_gh[t] / c_ah[a]);
      float lx = 0.5f * (sx - txv) * (sx - txv);
      float ly = 0.5f * (sy - tyv) * (sy - tyv);
      float lw = 0.5f * (in2 - twv) * (in2 - twv);
      float lh = 0.5f * (in3 - thv) * (in3 - thv);

      // tconf = IoU(gt, pred_at) stored for the conf kernel
      float px = sx + (float)gi, py = sy + (float)gj;
      float pw = expf(in2) * c_aw[a], ph = expf(in3) * c_ah[a];
      float tc = iou_box(s_gx[t], s_gy[t], s_gw[t], s_gh[t], px, py, pw, ph);
      tconf_ovr[(size_t)(b * NA + a) * HW + (size_t)gj * NW + gi] = tc;

      myloss += lx + ly + lw + lh + l_cls;
    }
    __syncthreads();
  }
  if (tid == 0) partT[b] = myloss;
}

__global__ __launch_bounds__(256) void kcells(const float* __restrict__ in,
                                              const float* __restrict__ gtbox,
                                              const int* __restrict__ nvalid,
                                              const float* __restrict__ tconf_ovr,
                                              float* __restrict__ partA) {
  __shared__ float s8[8];
  int idx = blockIdx.x * 256 + threadIdx.x;   // CELLS is an exact multiple of 256
  int b = idx / (NA * HW);
  int r = idx % (NA * HW);
  int a = r / HW;
  int ji = r % HW;
  int j = ji / NW, i = ji % NW;

  size_t base = (size_t)(b * NA + a) * CH * HW + ji;
  float in0 = in[base];
  float in1 = in[base + HW];
  float in2 = in[base + 2 * (size_t)HW];
  float in3 = in[base + 3 * (size_t)HW];
  float in4 = in[base + 4 * (size_t)HW];

  float sx = 1.0f / (1.0f + expf(-in0));
  float sy = 1.0f / (1.0f + expf(-in1));
  float conf = 1.0f / (1.0f + expf(-in4));
  float px = sx + (float)i, py = sy + (float)j;
  float pw = expf(in2) * c_aw[a], ph = expf(in3) * c_ah[a];

  int nv = nvalid[b];
  const float* g = gtbox + (size_t)b * MAXT * 4;
  float maxiou = 0.0f;
  for (int t = 0; t < nv; ++t) {
    float iou = iou_box(g[t * 4 + 0], g[t * 4 + 1], g[t * 4 + 2], g[t * 4 + 3],
                        px, py, pw, ph);
    maxiou = fmaxf(maxiou, iou);
  }

  float tov = tconf_ovr[idx];
  float contrib;
  if (tov >= 0.0f) {
    contrib = 2.5f * (conf - tov) * (conf - tov);        // OBJ_SCALE/2 * (conf-tconf)^2
  } else {
    float mask = (maxiou > 0.6f) ? 0.0f : 1.0f;          // NOOBJ_SCALE
    contrib = 0.5f * mask * conf * conf;                 // tconf = 0
  }

  float s = block_sum_256(contrib, s8);
  if (threadIdx.x == 0) partA[blockIdx.x] = s;
}

__global__ __launch_bounds__(256) void kfinal(const float* __restrict__ partT,
                                              const float* __restrict__ partA,
                                              float* __restrict__ out) {
  __shared__ float s8[8];
  int tid = threadIdx.x;
  float acc = 0.0f;
  for (int k = tid; k < NB; k += 256) acc += partT[k];
  for (int k = tid; k < NBLK_A; k += 256) acc += partA[k];
  float s = block_sum_256(acc, s8);
  if (tid == 0) out[0] = s;
}

extern "C" void kernel_launch(void* const* d_in, const int* in_sizes, int n_in,
                              void* d_out, int out_size, void* d_ws, size_t ws_size,
                              hipStream_t stream) {
  const float* inp = (const float*)d_in[0];   // (64, 425, 38, 38) f32
  const float* tgt = (const float*)d_in[1];   // (64, 250) f32
  float* out = (float*)d_out;                 // 1 f32

  char* ws = (char*)d_ws;
  float* tconf_ovr = (float*)ws;                                 // CELLS f32
  float* gtbox     = (float*)(ws + (size_t)CELLS * 4);           // NB*MAXT*4 f32
  int*   nvalid    = (int*)  (ws + (size_t)CELLS * 4 + (size_t)NB * MAXT * 16);
  float* partT     = (float*)((char*)nvalid + NB * 4);           // NB f32
  float* partA     = partT + NB;                                 // NBLK_A f32

  kinit<<<NBLK_A, 256, 0, stream>>>(tconf_ovr);
  ktargets<<<NB, 128, 0, stream>>>(inp, tgt, tconf_ovr, gtbox, nvalid, partT);
  kcells<<<NBLK_A, 256, 0, stream>>>(inp, gtbox, nvalid, tconf_ovr, partA);
  kfinal<<<1, 256, 0, stream>>>(partT, partA, out);
}